// RegionLayer_69045894250634
// MI455X (gfx1250) — compile-verified
//
#include <hip/hip_runtime.h>
#include <hip/hip_bf16.h>
#include <math.h>

#define NB    16
#define NA    5
#define NC    20
#define NH    64
#define NW    64
#define MAXB  50
#define PLANE (NH*NW)          // 4096
#define CELLS_PER_B (NA*PLANE) // 20480
#define BLK   256
#define BLKS_PER_B (CELLS_PER_B/BLK) // 80
#define NBLKS (NB*BLKS_PER_B)        // 1280
#define THRESH 0.6f
#define OBJECT_SCALE 5.0f

typedef __attribute__((ext_vector_type(2))) float v2f;
typedef __attribute__((ext_vector_type(8))) float v8f;

struct BoxRec {               // 48 bytes, 12 dwords
    float gx, gy, gw, gh, tlw, tlh;
    int   cls, bestn, gi, gj, valid, pad;
};

// ---------------- Kernel 1: per-box precompute (800 boxes) ----------------
__global__ void region_prep_boxes(const float* __restrict__ target,
                                  const float* __restrict__ anchors,
                                  BoxRec* __restrict__ boxes)
{
    int idx = blockIdx.x * blockDim.x + threadIdx.x;
    if (idx >= NB * MAXB) return;
    int b = idx / MAXB, t = idx % MAXB;
    const float* tb = target + (size_t)b * MAXB * 5;

    // validity = cumprod(x != 0) up to this box
    int valid = 1;
    for (int k = 0; k <= t; ++k) valid &= (tb[k*5 + 1] != 0.0f);

    float cls = tb[t*5 + 0];
    float gx  = tb[t*5 + 1] * (float)NW;
    float gy  = tb[t*5 + 2] * (float)NH;
    float gw  = tb[t*5 + 3] * (float)NW;
    float gh  = tb[t*5 + 4] * (float)NH;

    int gi = (int)gx; gi = min(max(gi, 0), NW - 1);
    int gj = (int)gy; gj = min(max(gj, 0), NH - 1);

    // best anchor by wh-IoU, first-max wins (matches jnp.argmax)
    int bestn = 0; float best = -1.0f;
    for (int a = 0; a < NA; ++a) {
        float aw = anchors[a*2 + 0], ah = anchors[a*2 + 1];
        float inter = fminf(gw, aw) * fminf(gh, ah);
        float uni   = gw*gh + aw*ah - inter;
        float s     = inter / uni;
        if (s > best) { best = s; bestn = a; }
    }

    BoxRec r;
    r.gx = gx; r.gy = gy; r.gw = gw; r.gh = gh;
    r.tlw = logf(gw / anchors[bestn*2 + 0]);
    r.tlh = logf(gh / anchors[bestn*2 + 1]);
    r.cls = (int)cls; r.bestn = bestn; r.gi = gi; r.gj = gj;
    r.valid = valid; r.pad = 0;
    boxes[idx] = r;
}

// ---------------- Kernel 2: fused per-cell loss + WMMA block reduce ----------------
__global__ void __launch_bounds__(BLK)
region_loss_main(const float* __restrict__ out,
                 const float* __restrict__ anchors,
                 const BoxRec* __restrict__ boxes,
                 float* __restrict__ partials)
{
    __shared__ BoxRec sb[MAXB];
    __shared__ float  waveSum[BLK/32];

    int blk = blockIdx.x;
    int b   = blk / BLKS_PER_B;
    int cell = (blk % BLKS_PER_B) * BLK + threadIdx.x;

    // cooperative copy of this image's 50 boxes into LDS (600 dwords)
    {
        const int* src = (const int*)(boxes + (size_t)b * MAXB);
        int* dst = (int*)sb;
        for (int t = threadIdx.x; t < MAXB * 12; t += BLK) dst[t] = src[t];
    }
    __syncthreads();

    int a   = cell >> 12;       // / 4096
    int rem = cell & 4095;      // j*64 + i
    int j   = rem >> 6, i = rem & 63;

    const float* base = out + ((size_t)b * (NA*25) + (size_t)a * 25) * PLANE + rem;

    float xo = base[0*PLANE];
    float yo = base[1*PLANE];
    float wo = base[2*PLANE];
    float ho = base[3*PLANE];
    float co = base[4*PLANE];

    float sx = 1.0f / (1.0f + expf(-xo));
    float sy = 1.0f / (1.0f + expf(-yo));
    float sc = 1.0f / (1.0f + expf(-co));

    float aw = anchors[a*2 + 0], ah = anchors[a*2 + 1];
    float px = sx + (float)i, py = sy + (float)j;
    float pw = expf(wo) * aw,  ph = expf(ho) * ah;

    // last valid GT box assigned to this (anchor, cell): last-writer-wins scan
    int assigned = -1;
    for (int t = 0; t < MAXB; ++t) {
        if (!sb[t].valid) break;                 // validity is a prefix property
        if (sb[t].bestn == a && sb[t].gi == i && sb[t].gj == j) assigned = t;
    }

    float tx = 0.5f, ty = 0.5f, tw = 0.0f, th = 0.0f, tconf = 0.0f;
    float cmask, lcls = 0.0f;

    if (assigned >= 0) {
        BoxRec r = sb[assigned];
        tx = r.gx - (float)i; ty = r.gy - (float)j; tw = r.tlw; th = r.tlh;
        // IoU(gt, predicted box at this cell)
        float L  = fmaxf(px - pw*0.5f, r.gx - r.gw*0.5f);
        float R  = fminf(px + pw*0.5f, r.gx + r.gw*0.5f);
        float T  = fmaxf(py - ph*0.5f, r.gy - r.gh*0.5f);
        float Bt = fminf(py + ph*0.5f, r.gy + r.gh*0.5f);
        float inter = fmaxf(R - L, 0.0f) * fmaxf(Bt - T, 0.0f);
        tconf = inter / (pw*ph + r.gw*r.gh - inter);
        cmask = OBJECT_SCALE;
        // cross-entropy over 20 class logits (two streaming passes, no spill array)
        const float* lg = base + 5*PLANE;
        float m = -1e30f;
        for (int c = 0; c < NC; ++c) m = fmaxf(m, lg[c*PLANE]);
        float s = 0.0f, lc = 0.0f;
        for (int c = 0; c < NC; ++c) {
            float v = lg[c*PLANE];
            s += expf(v - m);
            if (c == r.cls) lc = v;
        }
        lcls = (m + logf(s)) - lc;
    } else {
        // only the boolean (max IoU > 0.6) is needed -> early exit
        cmask = 1.0f;
        for (int t = 0; t < MAXB; ++t) {
            if (!sb[t].valid) break;
            float gx = sb[t].gx, gy = sb[t].gy, gw = sb[t].gw, gh = sb[t].gh;
            float L  = fmaxf(px - pw*0.5f, gx - gw*0.5f);
            float R  = fminf(px + pw*0.5f, gx + gw*0.5f);
            float T  = fmaxf(py - ph*0.5f, gy - gh*0.5f);
            float Bt = fminf(py + ph*0.5f, gy + gh*0.5f);
            float inter = fmaxf(R - L, 0.0f) * fmaxf(Bt - T, 0.0f);
            float iou = inter / (pw*ph + gw*gh - inter);
            if (iou > THRESH) { cmask = 0.0f; break; }
        }
    }

    float dx = sx - tx, dy = sy - ty, dw = wo - tw, dh = ho - th;
    float partial = 0.5f * (dx*dx + dy*dy + dw*dw + dh*dh);   // coord loss (mask==1)
    float dc = sc - tconf;
    partial += 0.5f * cmask * dc * dc;                         // conf loss
    partial += lcls;                                           // class loss

    // ---- wave32 reduction via V_WMMA_F32_16X16X4_F32: A(16x4)=partials, B=ones ----
    // A layout: lanes 0-15 -> {K0,K1}, lanes 16-31 -> {K2,K3}; put partial in K-even col.
    v2f A; A[0] = partial; A[1] = 0.0f;
    v2f Bv; Bv[0] = 1.0f;  Bv[1] = 1.0f;
    v8f acc = {};
    acc = __builtin_amdgcn_wmma_f32_16x16x4_f32(
        /*neg_a=*/false, A, /*neg_b=*/false, Bv,
        /*c_mod=*/(short)0, acc, /*reuse_a=*/false, /*reuse_b=*/false);
    // lane n<16 now holds rowsums of M=0..7 in acc[0..7]; lanes>=16 hold M=8..15
    float wsum = acc[0]+acc[1]+acc[2]+acc[3]+acc[4]+acc[5]+acc[6]+acc[7];
    wsum += __shfl_xor(wsum, 16);     // combine the two half-row groups

    int wid = threadIdx.x >> 5;
    if ((threadIdx.x & 31) == 0) waveSum[wid] = wsum;
    __syncthreads();
    if (threadIdx.x == 0) {
        float s = 0.0f;
        for (int w = 0; w < BLK/32; ++w) s += waveSum[w];   // fixed order: deterministic
        partials[blk] = s;
    }
}

// ---------------- Kernel 3: deterministic final reduction ----------------
__global__ void __launch_bounds__(256)
region_final_reduce(const float* __restrict__ partials, int n, float* __restrict__ out)
{
    __shared__ float sm[256];
    float s = 0.0f;
    for (int idx = threadIdx.x; idx < n; idx += 256) s += partials[idx]; // fixed order
    sm[threadIdx.x] = s;
    __syncthreads();
    for (int off = 128; off > 0; off >>= 1) {
        if (threadIdx.x < off) sm[threadIdx.x] += sm[threadIdx.x + off];
        __syncthreads();
    }
    if (threadIdx.x == 0) out[0] = sm[0];
}

extern "C" void kernel_launch(void* const* d_in, const int* in_sizes, int n_in,
                              void* d_out, int out_size, void* d_ws, size_t ws_size,
                              hipStream_t stream)
{
    const float* output  = (const float*)d_in[0]; // (16,125,64,64)
    const float* target  = (const float*)d_in[1]; // (16,250)
    const float* anchors = (const float*)d_in[2]; // (5,2)
    float* out = (float*)d_out;

    BoxRec* boxes   = (BoxRec*)d_ws;                              // 800*48 B
    float* partials = (float*)((char*)d_ws + 40960);              // 1280 floats

    region_prep_boxes<<<(NB*MAXB + 63)/64, 64, 0, stream>>>(target, anchors, boxes);
    region_loss_main<<<NBLKS, BLK, 0, stream>>>(output, anchors, boxes, partials);
    region_final_reduce<<<1, 256, 0, stream>>>(partials, NBLKS, out);
}